// FocalLossAdaptive_41128606826808
// MI455X (gfx1250) — compile-verified
//
#include <hip/hip_runtime.h>

// FocalLossAdaptive (FLSD) on MI455X / gfx1250.
// 524 MB single-pass stream -> HBM-bound floor ~22.5 us @ 23.3 TB/s.
// Online (running-max) softmax in base-2 domain, branchless 4-wide update,
// row staged through LDS with gfx1250 async copies (ASYNCcnt) when available.

typedef __attribute__((ext_vector_type(4))) float v4f;
typedef __attribute__((ext_vector_type(4))) int   v4i;

#define AS1 __attribute__((address_space(1)))
#define AS3 __attribute__((address_space(3)))

constexpr int   BLOCK  = 256;            // 8 waves (wave32)
constexpr int   WAVE   = 32;
constexpr int   NWAVES = BLOCK / WAVE;
constexpr int   DEPTH  = 4;              // async chunks in flight (4 KB each)
constexpr int   NBUF   = 8;              // ring slots (no reuse while readable)
constexpr float LOG2E  = 1.4426950408889634f;
constexpr float LN2    = 0.6931471805599453f;

#if __has_builtin(__builtin_amdgcn_global_load_async_to_lds_b128)
#define HAVE_ASYNC 1
#else
#define HAVE_ASYNC 0
#endif

#if __has_builtin(__builtin_amdgcn_s_wait_asynccnt)
#define WAIT_ASYNC(n) do { __builtin_amdgcn_s_wait_asynccnt(n); asm volatile("" ::: "memory"); } while (0)
#else
#define WAIT_ASYNC(n) asm volatile("s_wait_asynccnt %0" :: "i"(n) : "memory")
#endif

// Merge two (max, scaled-sum) pairs in base-2 domain.
__device__ __forceinline__ void merge_ms(float& m, float& s, float mo, float so) {
    float nm = fmaxf(m, mo);
    s = s * __builtin_amdgcn_exp2f(m - nm) + so * __builtin_amdgcn_exp2f(mo - nm);
    m = nm;
}

// Branchless 4-wide online update: 5 independent v_exp_f32 per 4 elements.
__device__ __forceinline__ void upd4(float& m, float& s, v4f v) {
    float y0 = v.x * LOG2E, y1 = v.y * LOG2E;
    float y2 = v.z * LOG2E, y3 = v.w * LOG2E;
    float lm = fmaxf(fmaxf(y0, y1), fmaxf(y2, y3));
    float nm = fmaxf(m, lm);
    s = s * __builtin_amdgcn_exp2f(m - nm)
      + __builtin_amdgcn_exp2f(y0 - nm)
      + __builtin_amdgcn_exp2f(y1 - nm)
      + __builtin_amdgcn_exp2f(y2 - nm)
      + __builtin_amdgcn_exp2f(y3 - nm);
    m = nm;
}

__global__ __launch_bounds__(BLOCK) void focal_row_kernel(
    const float* __restrict__ input,
    const int*   __restrict__ target,
    float*       __restrict__ row_loss,
    int C)
{
    const int row = blockIdx.x;
    const int tid = threadIdx.x;
    const float* __restrict__ rp  = input + (size_t)row * (size_t)C;
    const v4f*   __restrict__ rp4 = reinterpret_cast<const v4f*>(rp);
    const int n4   = C >> 2;            // 8000 float4s
    const int full = n4 / BLOCK;        // 31 uniform block-wide chunks

    float m = -3.0e38f;
    float s = 0.0f;

#if HAVE_ASYNC
    __shared__ v4f stage[NBUF * BLOCK];           // 32 KB ring
    if (full >= DEPTH) {
        // Prologue: put DEPTH chunks (16 KB/block) in flight on the async engine.
        #pragma unroll
        for (int c = 0; c < DEPTH; ++c)
            __builtin_amdgcn_global_load_async_to_lds_b128(
                (AS1 v4i*)(rp4 + c * BLOCK + tid),
                (AS3 v4i*)&stage[(c & (NBUF - 1)) * BLOCK + tid], 0, 0);
        // Steady state: wait oldest chunk, consume own wave's 512B slice, refill.
        for (int c = 0; c < full - DEPTH; ++c) {
            WAIT_ASYNC(DEPTH - 1);                 // chunk c landed (in-order)
            upd4(m, s, stage[(c & (NBUF - 1)) * BLOCK + tid]);
            __builtin_amdgcn_global_load_async_to_lds_b128(
                (AS1 v4i*)(rp4 + (c + DEPTH) * BLOCK + tid),
                (AS3 v4i*)&stage[((c + DEPTH) & (NBUF - 1)) * BLOCK + tid], 0, 0);
        }
        // Drain: literal immediates (builtin requires ICE).
        WAIT_ASYNC(3); upd4(m, s, stage[((full - 4) & (NBUF - 1)) * BLOCK + tid]);
        WAIT_ASYNC(2); upd4(m, s, stage[((full - 3) & (NBUF - 1)) * BLOCK + tid]);
        WAIT_ASYNC(1); upd4(m, s, stage[((full - 2) & (NBUF - 1)) * BLOCK + tid]);
        WAIT_ASYNC(0); upd4(m, s, stage[((full - 1) & (NBUF - 1)) * BLOCK + tid]);
    } else
#endif
    {
        // Direct NT-B128 stream with gfx1250 prefetch, uniform trip count.
        constexpr int PF_IT = 8;                   // prefetch 32 KB ahead
        const v4f* p = rp4 + tid;
        for (int it = 0; it < full; ++it) {
            if (it + PF_IT < full)
                __builtin_prefetch(p + PF_IT * BLOCK, 0, 1);  // global_prefetch_b8
            upd4(m, s, __builtin_nontemporal_load(p));
            p += BLOCK;
        }
    }

    // Tail: n4 - full*BLOCK = 64 float4s, first 64 lanes, one divergent step.
    const int done4 = full * BLOCK;
    if (tid < n4 - done4)
        upd4(m, s, __builtin_nontemporal_load(rp4 + done4 + tid));

    // wave32 butterfly reduction of (m, s).
    #pragma unroll
    for (int off = WAVE / 2; off >= 1; off >>= 1) {
        float mo = __shfl_xor(m, off, WAVE);
        float so = __shfl_xor(s, off, WAVE);
        merge_ms(m, s, mo, so);
    }

    // Cross-wave merge through LDS (8 waves per block).
    __shared__ float sm[NWAVES];
    __shared__ float ss[NWAVES];
    const int lane = tid & (WAVE - 1);
    const int wav  = tid >> 5;
    if (lane == 0) { sm[wav] = m; ss[wav] = s; }
    __syncthreads();

    if (tid == 0) {
        float M = sm[0], S = ss[0];
        #pragma unroll
        for (int w = 1; w < NWAVES; ++w) merge_ms(M, S, sm[w], ss[w]);

        const int   tgt = target[row];
        const float xt  = rp[tgt];
        const float t2  = xt * LOG2E - (M + __builtin_amdgcn_logf(S)); // v_log_f32
        const float logpt = t2 * LN2;
        const float pt    = __builtin_amdgcn_exp2f(t2);

        // gamma = 5 if pt < 0.2 else 3 -> integer powers.
        const float u  = 1.0f - pt;
        const float u3 = u * u * u;
        const float w  = (pt < 0.2f) ? (u3 * u * u) : u3;
        row_loss[row] = -w * logpt;
    }
}

// Deterministic tree reduction of per-row losses into d_out[0].
__global__ __launch_bounds__(BLOCK) void final_reduce_kernel(
    const float* __restrict__ row_loss,
    float*       __restrict__ out,
    int n)
{
    __shared__ float buf[BLOCK];
    float a = 0.0f;
    for (int i = threadIdx.x; i < n; i += BLOCK) a += row_loss[i];
    buf[threadIdx.x] = a;
    __syncthreads();
    #pragma unroll
    for (int off = BLOCK / 2; off >= 1; off >>= 1) {
        if (threadIdx.x < off) buf[threadIdx.x] += buf[threadIdx.x + off];
        __syncthreads();
    }
    if (threadIdx.x == 0) out[0] = buf[0];
}

extern "C" void kernel_launch(void* const* d_in, const int* in_sizes, int n_in,
                              void* d_out, int out_size, void* d_ws, size_t ws_size,
                              hipStream_t stream) {
    const float* input  = (const float*)d_in[0];
    const int*   target = (const int*)d_in[1];

    const int N = in_sizes[1];               // 4096 rows
    const int C = in_sizes[0] / N;           // 32000 classes

    float* row_loss = (float*)d_ws;          // N floats of scratch

    focal_row_kernel<<<N, BLOCK, 0, stream>>>(input, target, row_loss, C);
    final_reduce_kernel<<<1, BLOCK, 0, stream>>>(row_loss, (float*)d_out, N);
}